// Concat_26147760898611
// MI455X (gfx1250) — compile-verified
//
#include <hip/hip_runtime.h>

// CDNA5 / gfx1250 wave32 WMMA types
typedef __attribute__((ext_vector_type(2))) float v2f;   // A/B: 16x4 or 4x16 f32 = 2 VGPRs
typedef __attribute__((ext_vector_type(8))) float v8f;   // C/D: 16x16 f32 = 8 VGPRs

#define DIM     300
#define AW      2048      // article words
#define OW      64        // options words
#define CHUNKS  19        // ceil(300/16)
#define BATCH   256
#define WAVES_PER_BLOCK 8
#define NTASKS  (BATCH * CHUNKS)                    // 4864
#define NBLOCKS (NTASKS / WAVES_PER_BLOCK)          // 608 (exact)

// One wave per (batch, 16-dim chunk). Reduction over words done entirely in
// the matrix unit: D = ones(16x4) * B(4x16) + D  sums 4 words x 16 dims per
// v_wmma_f32_16x16x4_f32. Row M=0 of D holds the per-dim sums.
__global__ __launch_bounds__(256) void meanpool_wmma_kernel(
    const float* __restrict__ article,   // [256, 2048, 300]
    const float* __restrict__ options,   // [256,   64, 300]
    float* __restrict__ out)             // [256, 600]
{
    const int lane  = threadIdx.x & 31;
    const int wave  = threadIdx.x >> 5;
    const int task  = blockIdx.x * WAVES_PER_BLOCK + wave;  // 0 .. NTASKS-1
    const int chunk = task % CHUNKS;
    const int b     = task / CHUNKS;

    // B-matrix (4x16 f32, 2 VGPRs) lane mapping:
    //   lanes 0-15 : VGPR0 = K=0 row, VGPR1 = K=1 row, N = lane
    //   lanes 16-31: VGPR0 = K=2 row, VGPR1 = K=3 row, N = lane-16
    // (Any K permutation is harmless: A is all-ones, we sum over K.)
    const int n     = lane & 15;        // output dim within chunk (matches D's N)
    const int khalf = lane >> 4;        // which K pair this lane supplies
    const int dim   = chunk * 16 + n;
    const int dimc  = (dim < DIM) ? dim : (DIM - 1);  // clamp address only;
                                                      // cols >= 300 never stored

    v2f ones; ones.x = 1.0f; ones.y = 1.0f;           // A = ones(16x4)

    // ---------------- article: sum over 2048 words ----------------
    const float* pa = article + (size_t)b * AW * DIM + (size_t)(2 * khalf) * DIM + dimc;
    v8f a0 = {}; v8f a1 = {};
    for (int w = 0; w < AW; w += 8) {
        v2f b0, b1;
        b0.x = pa[0];        b0.y = pa[DIM];          // words w+2k, w+2k+1
        b1.x = pa[4 * DIM];  b1.y = pa[5 * DIM];      // words w+4+2k, w+5+2k
        a0 = __builtin_amdgcn_wmma_f32_16x16x4_f32(false, ones, false, b0,
                                                   (short)0, a0, false, false);
        a1 = __builtin_amdgcn_wmma_f32_16x16x4_f32(false, ones, false, b1,
                                                   (short)0, a1, false, false);
        pa += 8 * DIM;
    }

    // ---------------- options: sum over 64 words ----------------
    const float* po = options + (size_t)b * OW * DIM + (size_t)(2 * khalf) * DIM + dimc;
    v8f o0 = {}; v8f o1 = {};
    for (int w = 0; w < OW; w += 8) {
        v2f b0, b1;
        b0.x = po[0];        b0.y = po[DIM];
        b1.x = po[4 * DIM];  b1.y = po[5 * DIM];
        o0 = __builtin_amdgcn_wmma_f32_16x16x4_f32(false, ones, false, b0,
                                                   (short)0, o0, false, false);
        o1 = __builtin_amdgcn_wmma_f32_16x16x4_f32(false, ones, false, b1,
                                                   (short)0, o1, false, false);
        po += 8 * DIM;
    }

    // D layout: VGPR0, lanes 0-15 = row M=0, N=lane -> the column sums.
    v8f asum = a0 + a1;
    v8f osum = o0 + o1;
    if (lane < 16 && dim < DIM) {
        float* orow = out + (size_t)b * (2 * DIM);
        orow[dim]       = asum[0] * (1.0f / (float)AW);
        orow[DIM + dim] = osum[0] * (1.0f / (float)OW);
    }
}

extern "C" void kernel_launch(void* const* d_in, const int* in_sizes, int n_in,
                              void* d_out, int out_size, void* d_ws, size_t ws_size,
                              hipStream_t stream) {
    const float* article = (const float*)d_in[0];  // 256*2048*300 f32
    const float* options = (const float*)d_in[1];  // 256*64*300  f32
    float* out = (float*)d_out;                    // 256*600     f32
    (void)in_sizes; (void)n_in; (void)out_size; (void)d_ws; (void)ws_size;

    meanpool_wmma_kernel<<<NBLOCKS, 256, 0, stream>>>(article, options, out);
}